// BCE_sigmoid_negtive_bias_all_58119497450040
// MI455X (gfx1250) — compile-verified
//
#include <hip/hip_runtime.h>

// ---------------------------------------------------------------------------
// BCE-with-logits sum reduction for MI455X (gfx1250, wave32).
//   loss = sum( max(x,0) - x*t + log(1 + exp(-|x|)) ),  t = (label > 0)
// Memory-bound: ~184 MB streamed once -> NT b128 loads; V_WMMA_F32_16X16X4_F32
// as the wave-level reduction engine; deterministic two-pass final reduction.
// Hot loop uses 32-bit indices and hardware exp/log only.
// ---------------------------------------------------------------------------

typedef __attribute__((ext_vector_type(2))) float v2f;
typedef __attribute__((ext_vector_type(4))) float v4f;
typedef __attribute__((ext_vector_type(8))) float v8f;
typedef __attribute__((ext_vector_type(4))) int   v4i;

#define THREADS 256
#define BLOCKS  1024

__device__ __forceinline__ float bce_elem(float x, int l) {
    // stable softplus(x) - x*t:
    //   max(x,0) - (t ? x : 0) + log(1 + exp(-|x|)),  e = exp(-|x|) in (0,1]
    float e  = __expf(-fabsf(x));
    float xt = (l > 0) ? x : 0.0f;
    return fmaxf(x, 0.0f) - xt + __logf(1.0f + e);
}

// Wave32 sum via V_WMMA_F32_16X16X4_F32 with B = ones.
// A layout (16x4 f32): lane L<16 -> A[L][0..1] = {acc, 0};
//                      lane L>=16 -> A[L-16][2..3] = {acc, 0}.
// => D[M][N] = acc_M + acc_{M+16} for every N (B all-ones makes B layout moot).
// D layout: lane 0 VGPR r = D[r][0] (M=0..7), lane 16 VGPR r = D[8+r][0].
// So (sum of 8 D regs on lane 0) + (same on lane 16) = full wave sum.
__device__ __forceinline__ float wave_sum_wmma(float acc) {
    v2f a; a.x = acc;  a.y = 0.0f;
    v2f b; b.x = 1.0f; b.y = 1.0f;
    v8f c = {};
    c = __builtin_amdgcn_wmma_f32_16x16x4_f32(
            /*neg_a=*/false, a, /*neg_b=*/false, b,
            /*c_mod=*/(short)0, c, /*reuse_a=*/false, /*reuse_b=*/false);
    float s = c[0] + c[1] + c[2] + c[3] + c[4] + c[5] + c[6] + c[7];
    return __shfl(s, 0, 32) + __shfl(s, 16, 32);
}

// Deterministic block reduction; result valid on threadIdx.x == 0.
__device__ __forceinline__ float block_sum(float acc) {
    __shared__ float wsum[THREADS / 32];
    float w = wave_sum_wmma(acc);
    int lane = threadIdx.x & 31;
    int wid  = threadIdx.x >> 5;
    if (lane == 0) wsum[wid] = w;
    __syncthreads();
    float b = 0.0f;
    if (threadIdx.x == 0) {
        #pragma unroll
        for (int i = 0; i < THREADS / 32; ++i) b += wsum[i];
    }
    return b;
}

__device__ __forceinline__ float quad_bce(const v4f* __restrict__ x4,
                                          const v4i* __restrict__ l4, int i) {
    // streamed once -> non-temporal b128 loads (don't thrash 192MB L2)
    v4f xv = __builtin_nontemporal_load(&x4[i]);
    v4i lv = __builtin_nontemporal_load(&l4[i]);
    return bce_elem(xv.x, lv.x) + bce_elem(xv.y, lv.y) +
           bce_elem(xv.z, lv.z) + bce_elem(xv.w, lv.w);
}

__global__ void bce_partial_kernel(const float* __restrict__ x,
                                   const int*   __restrict__ labels,
                                   float*       __restrict__ partials,
                                   int n4) {
    const v4f* __restrict__ x4 = (const v4f*)x;
    const v4i* __restrict__ l4 = (const v4i*)labels;
    const int stride = BLOCKS * THREADS;
    int i = blockIdx.x * THREADS + threadIdx.x;

    float acc = 0.0f;
    // 2x unrolled grid-stride: 4 independent b128 loads in flight per step
    for (; i + stride < n4; i += 2 * stride) {
        acc += quad_bce(x4, l4, i);
        acc += quad_bce(x4, l4, i + stride);
    }
    if (i < n4) acc += quad_bce(x4, l4, i);

    float b = block_sum(acc);
    if (threadIdx.x == 0) partials[blockIdx.x] = b;
}

__global__ void bce_final_kernel(const float* __restrict__ partials, int nparts,
                                 const float* __restrict__ x,
                                 const int*   __restrict__ labels,
                                 int tail_start, int n,
                                 float* __restrict__ out) {
    float acc = 0.0f;
    for (int i = threadIdx.x; i < nparts; i += THREADS) acc += partials[i];
    float b = block_sum(acc);
    if (threadIdx.x == 0) {
        // scalar tail for n % 4 != 0 (zero-length for this problem size)
        for (int i = tail_start; i < n; ++i) b += bce_elem(x[i], labels[i]);
        out[0] = b;
    }
}

extern "C" void kernel_launch(void* const* d_in, const int* in_sizes, int n_in,
                              void* d_out, int out_size, void* d_ws, size_t ws_size,
                              hipStream_t stream) {
    const float* x      = (const float*)d_in[0];
    const int*   labels = (const int*)d_in[1];
    const int    n      = in_sizes[0];
    const int    n4     = n / 4;

    float* partials = (float*)d_ws;  // BLOCKS floats = 4 KB of scratch

    bce_partial_kernel<<<BLOCKS, THREADS, 0, stream>>>(x, labels, partials, n4);
    bce_final_kernel<<<1, THREADS, 0, stream>>>(partials, BLOCKS, x, labels,
                                                n4 * 4, n, (float*)d_out);
}